// CustomCTCLoss_27204322852899
// MI455X (gfx1250) — compile-verified
//
#include <hip/hip_runtime.h>
#include <stdint.h>

// ---------------- configuration (matches reference: N=64, C=64, T=4000, S=100) ----
#define CTC_C     64
#define CTC_TT    96           // timesteps per staged chunk
#define CTC_TTP   97           // padded row stride (odd -> conflict-free gathers)
#define CTC_PCOLS 14           // columns owned per producer wave (7 waves * 14 >= 96)
#define CTC_PER_LANE 7         // alpha slots per lane of wave 0 (32*7 = 224 >= L=201)
#define NEGV (-1.0e30f)
#define LOG2E 1.4426950408889634f
#define LN2_D 0.6931471805599453

// ---------------- CDNA5 async global->LDS copy --------------------------------
#if defined(__has_builtin)
# if __has_builtin(__builtin_amdgcn_global_load_async_to_lds_b32)
#  define CTC_HAVE_ASYNC 1
# endif
#endif
#ifndef CTC_HAVE_ASYNC
# define CTC_HAVE_ASYNC 0
#endif

#if defined(__has_builtin) && __has_builtin(__builtin_amdgcn_exp2f)
# define EXP2F(x) __builtin_amdgcn_exp2f(x)
#else
# define EXP2F(x) exp2f(x)
#endif
#if defined(__has_builtin) && __has_builtin(__builtin_amdgcn_logf)
# define LOG2F(x) __builtin_amdgcn_logf(x)   // v_log_f32 = log base 2
#else
# define LOG2F(x) log2f(x)
#endif

__device__ __forceinline__ void ctc_async_copy_f32(const float* g, float* l) {
#if CTC_HAVE_ASYNC
    __builtin_amdgcn_global_load_async_to_lds_b32((int*)g, (int*)l, 0, 0);
#else
    *l = *g;
#endif
}

__device__ __forceinline__ void ctc_async_wait0() {
#if CTC_HAVE_ASYNC
# if __has_builtin(__builtin_amdgcn_s_wait_asynccnt)
    __builtin_amdgcn_s_wait_asynccnt(0);
# else
    asm volatile("s_wait_asynccnt 0" ::: "memory");
# endif
#endif
}

// ---------------- producer: stage one [C x TT] tile (log2 units) ---------------
// Producer wave pw (0..6) owns columns [pw*PCOLS, pw*PCOLS+PCOLS): it async-loads
// ALL 64 class rows for its columns, rescales them to log2 units, and folds the
// per-column logZ2 into its private running sum (Kahan-free: double). Everything
// is wave-local; the only cross-wave sync is the chunk-boundary barrier.
__device__ __forceinline__ void ctc_stage_chunk(const float* __restrict__ gbase,
                                                int t0, int T, int Tn,
                                                float* __restrict__ tilebuf,
                                                int pw, int lane,
                                                double& zacc) {
    const int ttbase = pw * CTC_PCOLS;
    // async loads: 64 rows x PCOLS cols per wave, 32 lanes
    for (int i = lane; i < CTC_C * CTC_PCOLS; i += 32) {
        int c  = i / CTC_PCOLS;
        int tt = ttbase + (i - c * CTC_PCOLS);
        if (tt < CTC_TT) {
            int tg = t0 + tt;
            if (tg > T - 1) tg = T - 1;   // clamp (consumer never reads past steps)
            ctc_async_copy_f32(gbase + (size_t)c * T + tg, &tilebuf[c * CTC_TTP + tt]);
        }
    }
    ctc_async_wait0();
    // rescale this wave's own columns to log2 units (natural-log logits * log2(e))
    for (int i = lane; i < CTC_C * CTC_PCOLS; i += 32) {
        int c  = i / CTC_PCOLS;
        int tt = ttbase + (i - c * CTC_PCOLS);
        if (tt < CTC_TT) tilebuf[c * CTC_TTP + tt] *= LOG2E;
    }
    // per-column logZ2 = log2(sum_c 2^x) folded into the running normalizer sum
    int tt = ttbase + lane;
    if (lane < CTC_PCOLS && tt < CTC_TT) {
        float m = -3.4e38f;
        #pragma unroll 4
        for (int c = 0; c < CTC_C; ++c) m = fmaxf(m, tilebuf[c * CTC_TTP + tt]);
        float ssum = 0.f;
        #pragma unroll 4
        for (int c = 0; c < CTC_C; ++c) ssum += EXP2F(tilebuf[c * CTC_TTP + tt] - m);
        float lz = m + LOG2F(ssum);
        if (t0 + tt < Tn) zacc += (double)lz;   // only steps the recurrence consumes
    }
}

// ---------------- main kernel: one block per sample ----------------------------
__global__ __launch_bounds__(256, 1)
void ctc_alpha_kernel(const float* __restrict__ logits,
                      const int* __restrict__ targets,
                      const int* __restrict__ target_lengths,
                      const int* __restrict__ input_lengths,
                      float* __restrict__ loss_ws,
                      int T, int S) {
    __shared__ float  tile[2][CTC_C * CTC_TTP];   // double-buffered log2-logit tiles
    __shared__ float  alphaOut[32 * CTC_PER_LANE];
    __shared__ double zpart[7 * 32];              // per-producer-lane logZ sums

    const int n    = blockIdx.x;
    const int tid  = threadIdx.x;
    const int lane = tid & 31;
    const int wave = tid >> 5;

    const float* gbase = logits + (size_t)n * CTC_C * T;
    const int Tn  = input_lengths[n];
    const int nch = (Tn + CTC_TT - 1) / CTC_TT;
    const int L   = 2 * S + 1;

    // ---- recurrence-wave private state ----
    float a[CTC_PER_LANE];
    int   off[CTC_PER_LANE];    // tile row offset per slot: ext[s] * TTP
    bool  skip[CTC_PER_LANE];
    double zacc = 0.0;          // producers: running sum of logZ2 over consumed steps
    if (wave == 0) {
        #pragma unroll
        for (int j = 0; j < CTC_PER_LANE; ++j) {
            int s = lane * CTC_PER_LANE + j;
            int e = 0; bool sk = false;
            if (s < L && (s & 1)) {
                int idx = s >> 1;
                e = targets[n * S + idx];               // labels in [1, C)
                int ep = (idx > 0) ? targets[n * S + idx - 1] : 0;
                sk = (e != ep);                          // skip allowed iff label != prev label
            }
            off[j] = e * CTC_TTP; skip[j] = sk; a[j] = NEGV;
        }
    }

    // ---- prologue: stage chunk 0 ----
    if (wave > 0) ctc_stage_chunk(gbase, 0, T, Tn, tile[0], wave - 1, lane, zacc);
    __syncthreads();

    // ---- pipelined chunk loop ----
    for (int k = 0; k < nch; ++k) {
        const int cur = k & 1;
        if (wave == 0) {
            const int t0 = k * CTC_TT;
            int steps = Tn - t0; if (steps > CTC_TT) steps = CTC_TT;
            const float* tb = tile[cur];
            int tloc = 0;
            if (k == 0) {                                // alpha init at t = 0
                if (lane == 0) {
                    a[0] = tb[off[0]];                   // blank at s=0 (off[0]==0)
                    a[1] = tb[off[1]];                   // first label at s=1
                }
                tloc = 1;
            }
            for (; tloc < steps; ++tloc) {
                float up1 = __shfl_up(a[6], 1, 32);      // alpha[7*lane-1]
                float up2 = __shfl_up(a[5], 1, 32);      // alpha[7*lane-2]
                if (lane == 0) { up1 = NEGV; up2 = NEGV; }
                float na[CTC_PER_LANE];
                #pragma unroll
                for (int j = 0; j < CTC_PER_LANE; ++j) {
                    float am1 = (j == 0) ? up1 : a[j - 1];
                    float am2 = (j == 0) ? up2 : ((j == 1) ? up1 : a[j - 2]);
                    float a3  = skip[j] ? am2 : NEGV;
                    // lse3 with only 2 exponentials: the max term contributes 1.
                    float p = fmaxf(am1, a3), q = fminf(am1, a3);
                    float m = fmaxf(a[j], p), r = fminf(a[j], p);
                    float ss = 1.0f + EXP2F(r - m) + EXP2F(q - m);
                    na[j] = m + LOG2F(ss) + tb[off[j] + tloc];
                }
                #pragma unroll
                for (int j = 0; j < CTC_PER_LANE; ++j) a[j] = na[j];
            }
        } else if (k + 1 < nch) {
            ctc_stage_chunk(gbase, (k + 1) * CTC_TT, T, Tn,
                            tile[cur ^ 1], wave - 1, lane, zacc);
        }
        __syncthreads();
    }

    // ---- epilogue: per-sample loss ----
    if (wave == 0) {
        #pragma unroll
        for (int j = 0; j < CTC_PER_LANE; ++j)
            alphaOut[lane * CTC_PER_LANE + j] = a[j];
    } else {
        zpart[(wave - 1) * 32 + lane] = zacc;   // lanes that never accumulated write 0
    }
    __syncthreads();
    if (tid == 0) {
        double cz = 0.0;
        for (int i = 0; i < 7 * 32; ++i) cz += zpart[i];
        int tl = target_lengths[n];
        int i1 = 2 * tl;
        int i2 = i1 - 1; if (i2 < 0) i2 = 0;
        float v1 = alphaOut[i1], v2 = alphaOut[i2];
        float m  = fmaxf(v1, v2);
        float l2 = m + LOG2F(EXP2F(v1 - m) + EXP2F(v2 - m));   // lse2 in log2 units
        double loss = -(((double)l2 - cz) * LN2_D) / (double)tl;
        loss_ws[n] = (float)loss;
    }
}

// ---------------- finalize: mean over samples ----------------------------------
__global__ void ctc_finalize_kernel(const float* __restrict__ loss_ws,
                                    float* __restrict__ out, int N) {
    __shared__ float sh[64];
    int t = threadIdx.x;
    sh[t] = (t < N) ? loss_ws[t] : 0.f;
    __syncthreads();
    if (t == 0) {
        float s = 0.f;
        for (int i = 0; i < N; ++i) s += sh[i];
        out[0] = s / (float)N;
    }
}

// ---------------- host entry ----------------------------------------------------
extern "C" void kernel_launch(void* const* d_in, const int* in_sizes, int n_in,
                              void* d_out, int out_size, void* d_ws, size_t ws_size,
                              hipStream_t stream) {
    const float* logits = (const float*)d_in[0];   // (N, C, T) f32
    const int*   targets = (const int*)d_in[1];    // (N, S) i32
    const int*   tlen    = (const int*)d_in[2];    // (N,)   i32
    const int*   ilen    = (const int*)d_in[3];    // (N,)   i32
    float* out = (float*)d_out;
    float* ws  = (float*)d_ws;

    const int N = in_sizes[2];                 // 64
    const int S = in_sizes[1] / N;             // 100
    const int T = in_sizes[0] / (N * CTC_C);   // 4000

    ctc_alpha_kernel<<<N, 256, 0, stream>>>(logits, targets, tlen, ilen, ws, T, S);
    ctc_finalize_kernel<<<1, 64, 0, stream>>>(ws, out, N);
}